// LLA_41412074668179
// MI455X (gfx1250) — compile-verified
//
#include <hip/hip_runtime.h>
#include <cmath>
#include <cstdint>

// Problem constants (from reference): imgs [4,3,384,384] f32 -> out [4,1,48,384,384] f32
#define BATCH 4
#define CHN   3
#define HDIM  384
#define WDIM  384
#define HW    (HDIM * WDIM)
#define KTAP  48
#define HALO  24
#define TW    32
#define THT   16
#define LW    (TW + 2 * HALO)   // 80
#define LH    (THT + 2 * HALO)  // 64
#define NPIX  (LW * LH)         // 5120 pixels staged per block
#define NTHREADS (TW * THT)     // 512

struct PosConst { float v[KTAP]; };

__global__ __launch_bounds__(NTHREADS)
void LLA_41412074668179_kernel(const float* __restrict__ imgs,
                               float* __restrict__ out,
                               PosConst pc) {
    // Dynamic LDS: [pixel][4 floats] (xyz = channels, w = pad) => 81920 bytes
    extern __shared__ float4 smem4[];
    float* smem = (float*)smem4;

    const int tx  = threadIdx.x;
    const int ty  = threadIdx.y;
    const int tid = ty * TW + tx;
    const int x0  = blockIdx.x * TW - HALO;
    const int y0  = blockIdx.y * THT - HALO;
    const float* bimg = imgs + (size_t)blockIdx.z * CHN * HW;

    // Low 32 bits of a generic LDS pointer are the wave-relative LDS byte offset.
    const uint32_t lds_base = (uint32_t)(uintptr_t)smem;

    // ---- Stage tile (+halo, replicate-clamped) into LDS via CDNA5 async loads ----
    // Pixel-major: 10 iterations/thread, 3 async b32 gathers per pixel (one per channel).
    for (int pix = tid; pix < NPIX; pix += NTHREADS) {
        int ly = pix / LW;
        int lx = pix - ly * LW;
        int gy = y0 + ly; gy = gy < 0 ? 0 : (gy > HDIM - 1 ? HDIM - 1 : gy);
        int gx = x0 + lx; gx = gx < 0 ? 0 : (gx > WDIM - 1 ? WDIM - 1 : gx);
        uint32_t goff  = (uint32_t)(gy * WDIM + gx) * 4u;
        uint32_t laddr = lds_base + (uint32_t)pix * 16u;
        asm volatile("global_load_async_to_lds_b32 %0, %1, %2"
                     :: "v"(laddr), "v"(goff), "s"(bimg) : "memory");
        asm volatile("global_load_async_to_lds_b32 %0, %1, %2"
                     :: "v"(laddr + 4u), "v"(goff + (uint32_t)(HW * 4)), "s"(bimg) : "memory");
        asm volatile("global_load_async_to_lds_b32 %0, %1, %2"
                     :: "v"(laddr + 8u), "v"(goff + (uint32_t)(2 * HW * 4)), "s"(bimg) : "memory");
    }
    asm volatile("s_wait_asynccnt 0" ::: "memory");
    __syncthreads();

    // ---- Per-pixel compute ----
    const int p = (ty + HALO) * LW + (tx + HALO);
    float4 cv = smem4[p];
    const float c0 = cv.x, c1 = cv.y, c2 = cv.z;

    const int DIL[6] = {1, 2, 4, 8, 12, 24};
    // OFFSETS (row,col) order from reference, center excluded -> (dy,dx):
    const int DYt[8] = {-1, -1, -1,  0, 0,  1, 1, 1};
    const int DXt[8] = {-1,  0,  1, -1, 1, -1, 0, 1};

    // Pass 1: per-channel sum / sumsq over the 48 neighbors (for unbiased std).
    float sA0 = 0.f, sA1 = 0.f, sA2 = 0.f;
    float sB0 = 0.f, sB1 = 0.f, sB2 = 0.f;
#pragma unroll
    for (int di = 0; di < 6; ++di) {
        const int d = DIL[di];
#pragma unroll
        for (int t = 0; t < 8; ++t) {
            float4 nv = smem4[p + (DYt[t] * LW + DXt[t]) * d];
            sA0 += nv.x; sB0 = fmaf(nv.x, nv.x, sB0);
            sA1 += nv.y; sB1 = fmaf(nv.y, nv.y, sB1);
            sA2 += nv.z; sB2 = fmaf(nv.z, nv.z, sB2);
        }
    }
    // weight_c = log2(e)/3 / ((std_c + eps) * 0.3)^2  (log2-domain softmax)
    auto wgt = [](float s, float q) {
        float m   = s * (1.0f / 48.0f);
        float var = (q - 48.0f * m * m) * (1.0f / 47.0f);
        var = fmaxf(var, 0.0f);
        float t = (sqrtf(var) + 1e-8f) * 0.3f;
        return (1.4426950408889634f / 3.0f) / (t * t);
    };
    const float w0 = wgt(sA0, sB0);
    const float w1 = wgt(sA1, sB1);
    const float w2 = wgt(sA2, sB2);

    // Opaque memory clobber: prevents CSE of the LDS reads below with pass 1,
    // keeping the 144 neighbor values out of registers (occupancy > reg reuse;
    // the store stream needs many waves in flight to saturate HBM).
    asm volatile("" ::: "memory");

    // Pass 2: log2-domain affinities for all 48 taps (re-read from LDS, b128 each).
    float aff[KTAP];
#pragma unroll
    for (int di = 0; di < 6; ++di) {
        const int d = DIL[di];
#pragma unroll
        for (int t = 0; t < 8; ++t) {
            float4 nv = smem4[p + (DYt[t] * LW + DXt[t]) * d];
            float a0 = nv.x - c0, a1 = nv.y - c1, a2 = nv.z - c2;
            aff[di * 8 + t] = -fmaf(a0 * a0, w0, fmaf(a1 * a1, w1, a2 * a2 * w2));
        }
    }

    // Softmax over K (base-2 exponentials via native v_exp_f32; identical result
    // to base-e softmax since log2(e) was folded into the weights).
    float mx = aff[0];
#pragma unroll
    for (int k = 1; k < KTAP; ++k) mx = fmaxf(mx, aff[k]);
    float sum = 0.f;
#pragma unroll
    for (int k = 0; k < KTAP; ++k) {
        float e = __builtin_amdgcn_exp2f(aff[k] - mx);
        aff[k] = e;
        sum += e;
    }
    const float inv = 1.0f / sum;

    // Store: out[b, 0, k, gy, gx]; each k-store coalesces across the 32-wide x dim.
    // Nontemporal: output is write-once and never re-read; keep L2 for the input tile.
    const int gx = blockIdx.x * TW + tx;
    const int gy = blockIdx.y * THT + ty;
    const size_t obase = ((size_t)blockIdx.z * KTAP) * HW + (size_t)gy * WDIM + gx;
#pragma unroll
    for (int k = 0; k < KTAP; ++k)
        __builtin_nontemporal_store(fmaf(aff[k], inv, pc.v[k]),
                                    &out[obase + (size_t)k * HW]);
}

// Host-side: 0.01 * softmax(pos_aff) is a pure constant of the problem.
static void compute_pos_consts(float outv[KTAP]) {
    const double dil[6]  = {1, 2, 4, 8, 12, 24};
    const double sq2     = 1.4142135623730951;
    const double diag[8] = {sq2, 1, sq2, 1, 1, sq2, 1, sq2};
    double pos[KTAP];
    int k = 0;
    for (int d = 0; d < 6; ++d)
        for (int t = 0; t < 8; ++t) pos[k++] = diag[t] * dil[d];
    double m = 0;
    for (k = 0; k < KTAP; ++k) m += pos[k];
    m /= KTAP;
    double v = 0;
    for (k = 0; k < KTAP; ++k) { double e = pos[k] - m; v += e * e; }
    v /= (KTAP - 1);
    double sd = std::sqrt(v);
    double pa[KTAP], mx = -1e300;
    for (k = 0; k < KTAP; ++k) {
        double t = pos[k] / ((sd + 1e-8) * 0.3);
        pa[k] = -(t * t);
        if (pa[k] > mx) mx = pa[k];
    }
    double s = 0, e[KTAP];
    for (k = 0; k < KTAP; ++k) { e[k] = std::exp(pa[k] - mx); s += e[k]; }
    for (k = 0; k < KTAP; ++k) outv[k] = (float)(0.01 * e[k] / s);
}

extern "C" void kernel_launch(void* const* d_in, const int* in_sizes, int n_in,
                              void* d_out, int out_size, void* d_ws, size_t ws_size,
                              hipStream_t stream) {
    (void)in_sizes; (void)n_in; (void)out_size; (void)d_ws; (void)ws_size;
    const float* imgs = (const float*)d_in[0];
    float* out = (float*)d_out;

    PosConst pc;
    compute_pos_consts(pc.v);

    dim3 grid(WDIM / TW, HDIM / THT, BATCH);   // 12 x 24 x 4
    dim3 block(TW, THT, 1);                    // 512 threads = 16 wave32
    size_t shmem = (size_t)NPIX * 16;          // 81920 bytes of LDS
    LLA_41412074668179_kernel<<<grid, block, shmem, stream>>>(imgs, out, pc);
}